// PY_SYN_METAAS_23630910063285
// MI455X (gfx1250) — compile-verified
//
#include <hip/hip_runtime.h>
#include <hip/hip_bf16.h>

typedef float v2f __attribute__((ext_vector_type(2)));
typedef float v8f __attribute__((ext_vector_type(8)));

// ---------------- utility kernels ----------------
__global__ void k_zero(float* p, long long n) {
  long long i = (long long)blockIdx.x * blockDim.x + threadIdx.x;
  long long s = (long long)gridDim.x * blockDim.x;
  for (; i < n; i += s) p[i] = 0.f;
}

// dst[n*K + k] = src[k*Ncol + n]   (transpose KxN -> NxK)
__global__ void k_transpose(float* __restrict__ dst, const float* __restrict__ src,
                            int K, int Ncol) {
  int i = blockIdx.x * blockDim.x + threadIdx.x;
  if (i < K * Ncol) { int k = i / Ncol, n = i % Ncol; dst[n * K + k] = src[i]; }
}

// mean/var -> affine (alpha,beta):  BN(v) = v*alpha + beta
__global__ void k_fin(const float* __restrict__ sum, const float* __restrict__ sq,
                      const float* __restrict__ g, const float* __restrict__ be,
                      float* __restrict__ alpha, float* __restrict__ beta,
                      int F, float invE) {
  int j = threadIdx.x;
  if (j < F) {
    float m = sum[j] * invE;
    float v = sq[j] * invE - m * m;
    float a = g[j] * rsqrtf(v + 1e-5f);
    alpha[j] = a; beta[j] = be[j] - m * a;
  }
}

// ---------------- pass 1: BN1 stats + per-node edge counts ----------------
__global__ __launch_bounds__(256) void k_bn1_stats(
    const float* __restrict__ x, const float* __restrict__ ea,
    const int* __restrict__ eidx, const float* __restrict__ We1,
    const float* __restrict__ be1, float* __restrict__ bn1_sum,
    float* __restrict__ bn1_sq, float* __restrict__ cnt, int E) {
  __shared__ float sW[9 * 84];
  __shared__ float sb[84], ssum[84], ssq[84];
  int t = threadIdx.x;
  for (int i = t; i < 9 * 84; i += 256) sW[i] = We1[i];
  for (int i = t; i < 84; i += 256) { sb[i] = be1[i]; ssum[i] = 0.f; ssq[i] = 0.f; }
  __syncthreads();
  int e = blockIdx.x * 256 + t;
  bool valid = e < E;
  float f[9];
  if (valid) {
    int r = eidx[e], c = eidx[E + e];
    float4 xr = ((const float4*)x)[r], xc = ((const float4*)x)[c];
    f[0]=xr.x; f[1]=xr.y; f[2]=xr.z; f[3]=xr.w;
    f[4]=xc.x; f[5]=xc.y; f[6]=xc.z; f[7]=xc.w;
    f[8]=ea[e];
    atomicAdd(&cnt[r], 1.0f);
  } else {
    #pragma unroll
    for (int i = 0; i < 9; ++i) f[i] = 0.f;
  }
  for (int j = 0; j < 84; ++j) {
    const float* w = &sW[j];
    float v = valid ? sb[j] : 0.f;
    v += f[0]*w[0] + f[1]*w[84] + f[2]*w[168] + f[3]*w[252] + f[4]*w[336]
       + f[5]*w[420] + f[6]*w[504] + f[7]*w[588] + f[8]*w[672];
    float v2 = v * v;
    #pragma unroll
    for (int off = 16; off; off >>= 1) {
      v  += __shfl_xor(v,  off, 32);
      v2 += __shfl_xor(v2, off, 32);
    }
    if ((t & 31) == 0) { atomicAdd(&ssum[j], v); atomicAdd(&ssq[j], v2); }
  }
  __syncthreads();
  for (int i = t; i < 84; i += 256) {
    atomicAdd(&bn1_sum[i], ssum[i]);
    atomicAdd(&bn1_sq[i],  ssq[i]);
  }
}

// ---------------- pass 2: edge GEMM (WMMA f32 16x16x4), scatter + BN2 stats ----
#define AS 90  // LDS A-tile row stride (floats), even for b64 loads

__global__ __launch_bounds__(256) void k_edge_gemm(
    const float* __restrict__ x, const float* __restrict__ ea,
    const int* __restrict__ eidx, const float* __restrict__ We1,
    const float* __restrict__ be1, const float* __restrict__ alpha1,
    const float* __restrict__ beta1, const float* __restrict__ Wn1T,
    const float* __restrict__ bn1, float* __restrict__ agg,
    float* __restrict__ bn2_sum, float* __restrict__ bn2_sq,
    int E, int ntiles) {
  __shared__ float sW[9 * 84];
  __shared__ float sca[84], scb[84];
  __shared__ __align__(16) float sA[8][16 * AS];
  __shared__ int sR[8][16];
  __shared__ float sSum[128], sSq[128];
  int t = threadIdx.x, lane = t & 31, wv = t >> 5;
  int m = lane & 15, hf = lane >> 4;
  for (int i = t; i < 9 * 84; i += 256) sW[i] = We1[i];
  for (int i = t; i < 84; i += 256) {
    float a = alpha1[i]; sca[i] = a; scb[i] = beta1[i] + be1[i] * a;
  }
  for (int i = t; i < 128; i += 256) { sSum[i] = 0.f; sSq[i] = 0.f; }
  __syncthreads();
  float* A = sA[wv];
  for (int tile = blockIdx.x * 8 + wv; tile < ntiles; tile += gridDim.x * 8) {
    int e = tile * 16 + m;
    int r = eidx[e], c = eidx[E + e];
    float4 xr = ((const float4*)x)[r], xc = ((const float4*)x)[c];
    float f0=xr.x,f1=xr.y,f2=xr.z,f3=xr.w,f4=xc.x,f5=xc.y,f6=xc.z,f7=xc.w,f8=ea[e];
    if (hf == 0) {
      A[m*AS+0]=f4; A[m*AS+1]=f5; A[m*AS+2]=f6; A[m*AS+3]=f7;
      sR[wv][m] = r;
    }
    for (int j = hf * 42; j < hf * 42 + 42; ++j) {   // e1 = BN1(e_in @ We1 + b)
      const float* w = &sW[j];
      float v = f0*w[0] + f1*w[84] + f2*w[168] + f3*w[252] + f4*w[336]
              + f5*w[420] + f6*w[504] + f7*w[588] + f8*w[672];
      A[m*AS + 4 + j] = v * sca[j] + scb[j];
    }
    asm volatile("s_wait_dscnt 0x0" ::: "memory");   // intra-wave LDS RAW fence

    v8f acc[8];
    v8f z8 = {0.f,0.f,0.f,0.f,0.f,0.f,0.f,0.f};
    #pragma unroll
    for (int nt = 0; nt < 8; ++nt) acc[nt] = z8;
    for (int ks = 0; ks < 22; ++ks) {                // K = 88 = 22 x 4
      int k0 = ks * 4 + hf * 2;
      v2f a = *(const v2f*)&A[m * AS + k0];
      #pragma unroll
      for (int nt = 0; nt < 8; ++nt) {
        int n = nt * 16 + m;
        v2f b = *(const v2f*)&Wn1T[n * 88 + k0];
        acc[nt] = __builtin_amdgcn_wmma_f32_16x16x4_f32(
            false, a, false, b, (short)0, acc[nt], false, false);
      }
    }
    #pragma unroll
    for (int nt = 0; nt < 8; ++nt) {
      int n = nt * 16 + m;
      float bias = bn1[n];
      float s = 0.f, q = 0.f;
      #pragma unroll
      for (int v = 0; v < 8; ++v) {
        float h = acc[nt][v] + bias;                 // h_pre (pre-BN2)
        acc[nt][v] = h;
        s += h; q += h * h;
      }
      atomicAdd(&sSum[n], s);
      atomicAdd(&sSq[n],  q);
      #pragma unroll
      for (int v = 0; v < 8; ++v) {                  // scatter h_pre -> agg[row]
        int rr = sR[wv][v + hf * 8];
        atomicAdd(&agg[(long long)rr * 128 + n], acc[nt][v]);
      }
    }
  }
  __syncthreads();
  for (int i = t; i < 128; i += 256) {
    atomicAdd(&bn2_sum[i], sSum[i]);
    atomicAdd(&bn2_sq[i],  sSq[i]);
  }
}

// ---------------- pass 3: node MLP (WMMA), x1 = relu([x,aggBN] @ Wm2 + b) -----
#define ASN 134  // 132 rounded to even stride

__global__ __launch_bounds__(128) void k_node(
    const float* __restrict__ x, const float* __restrict__ agg,
    const float* __restrict__ cnt, const float* __restrict__ alpha2,
    const float* __restrict__ beta2, const float* __restrict__ Wm2T,
    const float* __restrict__ bm2, float* __restrict__ x1, int ntiles) {
  __shared__ __align__(16) float sA[4][16 * ASN];
  int t = threadIdx.x, lane = t & 31, wv = t >> 5;
  int m = lane & 15, hf = lane >> 4;
  int tile = blockIdx.x * 4 + wv;
  if (tile >= ntiles) return;                         // no barriers below
  float* A = sA[wv];
  int nbase = tile * 16;
  int node = nbase + m;
  if (hf == 0) {
    float4 xv = ((const float4*)x)[node];
    A[m*ASN+0]=xv.x; A[m*ASN+1]=xv.y; A[m*ASN+2]=xv.z; A[m*ASN+3]=xv.w;
  }
  float c = cnt[node];
  float inv = c > 0.f ? 1.f / c : 0.f;
  const float* ag = &agg[(long long)node * 128];
  for (int f = hf * 64; f < hf * 64 + 64; ++f) {      // BN2-affine of scatter_mean
    float v = (c > 0.f) ? ag[f] * inv * alpha2[f] + beta2[f] : 0.f;
    A[m*ASN + 4 + f] = v;
  }
  asm volatile("s_wait_dscnt 0x0" ::: "memory");

  v8f acc[4];
  v8f z8 = {0.f,0.f,0.f,0.f,0.f,0.f,0.f,0.f};
  #pragma unroll
  for (int nt = 0; nt < 4; ++nt) acc[nt] = z8;
  for (int ks = 0; ks < 33; ++ks) {                   // K = 132 = 33 x 4
    int k0 = ks * 4 + hf * 2;
    v2f a = *(const v2f*)&A[m * ASN + k0];
    #pragma unroll
    for (int nt = 0; nt < 4; ++nt) {
      int n = nt * 16 + m;
      v2f b = *(const v2f*)&Wm2T[n * 132 + k0];
      acc[nt] = __builtin_amdgcn_wmma_f32_16x16x4_f32(
          false, a, false, b, (short)0, acc[nt], false, false);
    }
  }
  #pragma unroll
  for (int nt = 0; nt < 4; ++nt) {
    int n = nt * 16 + m;
    float bias = bm2[n];
    #pragma unroll
    for (int v = 0; v < 8; ++v) {
      int nd = nbase + v + hf * 8;
      x1[(long long)nd * 64 + n] = fmaxf(acc[nt][v] + bias, 0.f);
    }
  }
}

// ---------------- pass 4: e2 = relu([x1[row], x1[col], e1] @ We2 + b) ---------
__global__ __launch_bounds__(256) void k_e2(
    const float* __restrict__ x, const float* __restrict__ ea,
    const int* __restrict__ eidx, const float* __restrict__ We1,
    const float* __restrict__ be1, const float* __restrict__ alpha1,
    const float* __restrict__ beta1, const float* __restrict__ x1,
    const float* __restrict__ We2, const float* __restrict__ be2,
    float* __restrict__ out, int E) {
  __shared__ float sW[9 * 84];
  __shared__ float sca[84], scb[84], sW2[212];
  int t = threadIdx.x;
  for (int i = t; i < 9 * 84; i += 256) sW[i] = We1[i];
  for (int i = t; i < 84; i += 256) {
    float a = alpha1[i]; sca[i] = a; scb[i] = beta1[i] + be1[i] * a;
  }
  for (int i = t; i < 212; i += 256) sW2[i] = We2[i];
  __syncthreads();
  int e = blockIdx.x * 256 + t;
  if (e >= E) return;
  int r = eidx[e], c = eidx[E + e];
  float acc = be2[0];
  const float4* xr1 = (const float4*)&x1[(long long)r * 64];
  const float4* xc1 = (const float4*)&x1[(long long)c * 64];
  #pragma unroll
  for (int i = 0; i < 16; ++i) {
    float4 v = xr1[i];
    acc += v.x*sW2[i*4] + v.y*sW2[i*4+1] + v.z*sW2[i*4+2] + v.w*sW2[i*4+3];
  }
  #pragma unroll
  for (int i = 0; i < 16; ++i) {
    float4 v = xc1[i];
    acc += v.x*sW2[64+i*4] + v.y*sW2[64+i*4+1] + v.z*sW2[64+i*4+2] + v.w*sW2[64+i*4+3];
  }
  float4 xr = ((const float4*)x)[r], xc = ((const float4*)x)[c];
  float f0=xr.x,f1=xr.y,f2=xr.z,f3=xr.w,f4=xc.x,f5=xc.y,f6=xc.z,f7=xc.w,f8=ea[e];
  for (int j = 0; j < 84; ++j) {                      // recompute e1 on the fly
    const float* w = &sW[j];
    float v = f0*w[0] + f1*w[84] + f2*w[168] + f3*w[252] + f4*w[336]
            + f5*w[420] + f6*w[504] + f7*w[588] + f8*w[672];
    acc += (v * sca[j] + scb[j]) * sW2[128 + j];
  }
  out[e] = fmaxf(acc, 0.f);
}

// ---------------- launcher ----------------
extern "C" void kernel_launch(void* const* d_in, const int* in_sizes, int n_in,
                              void* d_out, int out_size, void* d_ws, size_t ws_size,
                              hipStream_t stream) {
  const float* x    = (const float*)d_in[0];
  const float* ea   = (const float*)d_in[1];
  const int*   eidx = (const int*)d_in[2];
  const float* We1  = (const float*)d_in[3];
  const float* be1  = (const float*)d_in[4];
  const float* gbn1 = (const float*)d_in[5];
  const float* bebn1= (const float*)d_in[6];
  const float* Wn1  = (const float*)d_in[7];
  const float* bn1  = (const float*)d_in[8];
  const float* gbn2 = (const float*)d_in[9];
  const float* bebn2= (const float*)d_in[10];
  const float* Wm2  = (const float*)d_in[11];
  const float* bm2  = (const float*)d_in[12];
  const float* We2  = (const float*)d_in[13];
  const float* be2  = (const float*)d_in[14];
  int N = in_sizes[0] / 4;
  int E = in_sizes[1];

  float* ws = (float*)d_ws;
  long long OFF_X1  = (long long)N * 128;
  long long OFF_CNT = (long long)N * 192;
  long long OFF_ST  = (long long)N * 193;
  float* agg     = ws;
  float* x1      = ws + OFF_X1;
  float* cnt     = ws + OFF_CNT;
  float* st      = ws + OFF_ST;
  float* bn1_sum = st + 0,   *bn1_sq = st + 96,  *alpha1 = st + 192, *beta1 = st + 288;
  float* bn2_sum = st + 384, *bn2_sq = st + 512, *alpha2 = st + 640, *beta2 = st + 768;
  float* Wn1T    = st + 1024;          // [128][88]
  float* Wm2T    = Wn1T + 128 * 88;    // [64][132]

  k_zero<<<2048, 256, 0, stream>>>(agg, (long long)N * 128);
  k_zero<<<256, 256, 0, stream>>>(cnt, (long long)N + 1024);
  k_transpose<<<(88 * 128 + 255) / 256, 256, 0, stream>>>(Wn1T, Wn1, 88, 128);
  k_transpose<<<(132 * 64 + 255) / 256, 256, 0, stream>>>(Wm2T, Wm2, 132, 64);

  k_bn1_stats<<<(E + 255) / 256, 256, 0, stream>>>(x, ea, eidx, We1, be1,
                                                   bn1_sum, bn1_sq, cnt, E);
  k_fin<<<1, 128, 0, stream>>>(bn1_sum, bn1_sq, gbn1, bebn1, alpha1, beta1, 84, 1.0f / E);

  int etiles = E / 16;
  k_edge_gemm<<<(etiles + 7) / 8, 256, 0, stream>>>(x, ea, eidx, We1, be1,
                                                    alpha1, beta1, Wn1T, bn1,
                                                    agg, bn2_sum, bn2_sq, E, etiles);
  k_fin<<<1, 128, 0, stream>>>(bn2_sum, bn2_sq, gbn2, bebn2, alpha2, beta2, 128, 1.0f / E);

  int ntile = N / 16;
  k_node<<<(ntile + 3) / 4, 128, 0, stream>>>(x, agg, cnt, alpha2, beta2,
                                              Wm2T, bm2, x1, ntile);

  k_e2<<<(E + 255) / 256, 256, 0, stream>>>(x, ea, eidx, We1, be1, alpha1, beta1,
                                            x1, We2, be2, (float*)d_out, E);
}